// HybridPooler_86234353369524
// MI455X (gfx1250) — compile-verified
//
#include <hip/hip_runtime.h>
#include <math.h>

// ---------------- constants (match reference) ----------------
constexpr int Dd = 768;     // feature dim
constexpr int Ss = 2048;    // sequence length (after clf)
constexpr int Bb = 16;      // batch
constexpr int Mm = 8;       // queries
constexpr int NCHUNK = 16;  // S chunks for pooling pass
constexpr int CHUNK = 128;  // tokens per chunk
constexpr int NSEG = 8;     // S segments for merged pass

typedef __attribute__((ext_vector_type(2))) float v2f;
typedef __attribute__((ext_vector_type(8))) float v8f;

// ---------------- helpers ----------------
__device__ __forceinline__ float wave_sum(float v) {
#pragma unroll
  for (int off = 16; off > 0; off >>= 1) v += __shfl_xor(v, off, 32);
  return v;
}

__device__ __forceinline__ float gelu_exact(float x) {
  return 0.5f * x * (1.0f + erff(x * 0.70710678118654752440f));
}

// =====================================================================
// Kernel 1: fused masked pooling partials + LayerNorm + q-scores.
// grid (NCHUNK, B), block 256 (8 waves). Each wave owns one token at a
// time; each lane owns 24 columns (6 x float4). Pooling partials are
// combined across the 8 waves in LDS and written per-(b,chunk) — no
// atomics, fully deterministic.
// =====================================================================
__global__ __launch_bounds__(256) void k_pool_scores(
    const float* __restrict__ tokens, const long long* __restrict__ lengths,
    const float* __restrict__ q, const float* __restrict__ lns,
    const float* __restrict__ lnb, float* __restrict__ scores,
    float* __restrict__ psum, float* __restrict__ pmax,
    float* __restrict__ pmin) {
  __shared__ float q_s[Mm * Dd];  // 24 KB
  __shared__ float sh[8 * Dd];    // 24 KB
  const int tid = threadIdx.x;
  const int lane = tid & 31;
  const int w = tid >> 5;
  const int b = blockIdx.y;
  const int chunk = blockIdx.x;

  for (int i = tid; i < Mm * Dd; i += 256) q_s[i] = q[i];
  __syncthreads();

  float sc[24], bi[24];
#pragma unroll
  for (int j = 0; j < 6; ++j) {
    const int d0 = j * 128 + lane * 4;
    float4 s4 = *(const float4*)(lns + d0);
    float4 b4 = *(const float4*)(lnb + d0);
    sc[j * 4 + 0] = s4.x; sc[j * 4 + 1] = s4.y; sc[j * 4 + 2] = s4.z; sc[j * 4 + 3] = s4.w;
    bi[j * 4 + 0] = b4.x; bi[j * 4 + 1] = b4.y; bi[j * 4 + 2] = b4.z; bi[j * 4 + 3] = b4.w;
  }

  float accS[24], accMx[24], accMn[24];
#pragma unroll
  for (int j = 0; j < 24; ++j) {
    accS[j] = 0.f; accMx[j] = -INFINITY; accMn[j] = INFINITY;
  }

  const int len = (int)lengths[b];
  const float invD = 1.0f / (float)Dd;

  for (int i = 0; i < 16; ++i) {
    const int s = chunk * CHUNK + w + 8 * i;  // wave-uniform token index
    if (s >= len) continue;                   // masked: no pooling, no score
    const float* xp = tokens + ((size_t)b * (Ss + 1) + 1 + (size_t)s) * Dd;
    __builtin_prefetch(xp + 8 * Dd, 0, 0);    // next token for this wave
    float xv[24];
#pragma unroll
    for (int j = 0; j < 6; ++j) {
      float4 v4 = *(const float4*)(xp + j * 128 + lane * 4);
      xv[j * 4 + 0] = v4.x; xv[j * 4 + 1] = v4.y;
      xv[j * 4 + 2] = v4.z; xv[j * 4 + 3] = v4.w;
    }
    float ps = 0.f, pq = 0.f;
#pragma unroll
    for (int j = 0; j < 24; ++j) {
      const float v = xv[j];
      accS[j] += v;
      accMx[j] = fmaxf(accMx[j], v);
      accMn[j] = fminf(accMn[j], v);
      ps += v;
      pq = fmaf(v, v, pq);
    }
    ps = wave_sum(ps);
    pq = wave_sum(pq);
    const float mu = ps * invD;
    const float var = fmaxf(pq * invD - mu * mu, 0.f);
    const float rstd = rsqrtf(var + 1e-5f);
#pragma unroll
    for (int j = 0; j < 24; ++j) xv[j] = (xv[j] - mu) * rstd * sc[j] + bi[j];

    float sval = 0.f;
#pragma unroll
    for (int m = 0; m < Mm; ++m) {
      float dot = 0.f;
#pragma unroll
      for (int j = 0; j < 24; ++j) {
        const int d0 = (j >> 2) * 128 + lane * 4 + (j & 3);
        dot = fmaf(q_s[m * Dd + d0], xv[j], dot);
      }
      dot = wave_sum(dot);
      if (lane == m) sval = dot;
    }
    if (lane < Mm)
      scores[((size_t)b * Mm + lane) * Ss + s] = sval * 0.03608439182435161f;  // 1/sqrt(768)
  }

  // ---- cross-wave combine of pooling partials (sum, max, min) ----
  const size_t pbase = ((size_t)b * NCHUNK + chunk) * Dd;
  __syncthreads();
#pragma unroll
  for (int j = 0; j < 24; ++j)
    sh[w * Dd + (j >> 2) * 128 + lane * 4 + (j & 3)] = accS[j];
  __syncthreads();
  for (int r = 0; r < 3; ++r) {
    const int col = tid + r * 256;
    float v = sh[col];
#pragma unroll
    for (int w2 = 1; w2 < 8; ++w2) v += sh[w2 * Dd + col];
    psum[pbase + col] = v;
  }
  __syncthreads();
#pragma unroll
  for (int j = 0; j < 24; ++j)
    sh[w * Dd + (j >> 2) * 128 + lane * 4 + (j & 3)] = accMx[j];
  __syncthreads();
  for (int r = 0; r < 3; ++r) {
    const int col = tid + r * 256;
    float v = sh[col];
#pragma unroll
    for (int w2 = 1; w2 < 8; ++w2) v = fmaxf(v, sh[w2 * Dd + col]);
    pmax[pbase + col] = v;
  }
  __syncthreads();
#pragma unroll
  for (int j = 0; j < 24; ++j)
    sh[w * Dd + (j >> 2) * 128 + lane * 4 + (j & 3)] = accMn[j];
  __syncthreads();
  for (int r = 0; r < 3; ++r) {
    const int col = tid + r * 256;
    float v = sh[col];
#pragma unroll
    for (int w2 = 1; w2 < 8; ++w2) v = fminf(v, sh[w2 * Dd + col]);
    pmin[pbase + col] = v;
  }
}

// =====================================================================
// Kernel 2: reduce per-chunk partials -> pooled_trad (B, 3*D)
// =====================================================================
__global__ __launch_bounds__(256) void k_reduce_pool(
    const float* __restrict__ psum, const float* __restrict__ pmax,
    const float* __restrict__ pmin, const long long* __restrict__ lengths,
    float* __restrict__ pooled_trad) {
  const int b = blockIdx.x;
  const int tid = threadIdx.x;
  const float inv_len = 1.0f / (float)lengths[b];
  for (int r = 0; r < 3; ++r) {
    const int col = tid + r * 256;
    float s = 0.f, mx = -INFINITY, mn = INFINITY;
    for (int c = 0; c < NCHUNK; ++c) {
      const size_t idx = ((size_t)b * NCHUNK + c) * Dd + col;
      s += psum[idx];
      mx = fmaxf(mx, pmax[idx]);
      mn = fminf(mn, pmin[idx]);
    }
    pooled_trad[(size_t)b * (3 * Dd) + col] = s * inv_len;
    pooled_trad[(size_t)b * (3 * Dd) + Dd + col] = mx;
    pooled_trad[(size_t)b * (3 * Dd) + 2 * Dd + col] = mn;
  }
}

// =====================================================================
// Kernel 3: masked softmax over S, in place on scores (B, M, S)
// =====================================================================
__global__ __launch_bounds__(256) void k_softmax(
    float* __restrict__ scores, const long long* __restrict__ lengths) {
  __shared__ float red[256];
  const int row = blockIdx.x;  // b*M + m
  const int b = row >> 3;
  const int tid = threadIdx.x;
  const int len = (int)lengths[b];
  float* __restrict__ p = scores + (size_t)row * Ss;

  float mx = -INFINITY;
  for (int s = tid; s < len; s += 256) mx = fmaxf(mx, p[s]);
  red[tid] = mx;
  __syncthreads();
  for (int o = 128; o > 0; o >>= 1) {
    if (tid < o) red[tid] = fmaxf(red[tid], red[tid + o]);
    __syncthreads();
  }
  const float bmx = red[0];
  __syncthreads();

  float sum = 0.f;
  for (int s = tid; s < len; s += 256) sum += expf(p[s] - bmx);
  red[tid] = sum;
  __syncthreads();
  for (int o = 128; o > 0; o >>= 1) {
    if (tid < o) red[tid] += red[tid + o];
    __syncthreads();
  }
  const float inv = 1.0f / red[0];
  __syncthreads();

  for (int s = tid; s < Ss; s += 256)
    p[s] = (s < len) ? expf(p[s] - bmx) * inv : 0.f;
}

// =====================================================================
// Kernel 4: copy clf token into tail of pooled_learn (B, (M+1)*D)
// =====================================================================
__global__ __launch_bounds__(256) void k_clf_copy(
    const float* __restrict__ tokens, float* __restrict__ pooled_learn) {
  const int b = blockIdx.x;
  const int tid = threadIdx.x;
  for (int r = 0; r < 3; ++r) {
    const int col = tid + r * 256;
    pooled_learn[(size_t)b * ((Mm + 1) * Dd) + Mm * Dd + col] =
        tokens[(size_t)b * (Ss + 1) * Dd + col];
  }
}

// =====================================================================
// Kernel 5a: merged partials. S is split into NSEG segments for
// parallelism; each block owns (128 cols) x (S/NSEG tokens) and writes a
// deterministic per-segment partial.  x comes from L2 (first pass warmed
// it). grid (D/128, NSEG, B), block 128.
// =====================================================================
__global__ __launch_bounds__(128) void k_merged(
    const float* __restrict__ tokens, const float* __restrict__ attn,
    float* __restrict__ merged_part) {
  __shared__ float att[Mm * 128];
  const int b = blockIdx.z;
  const int seg = blockIdx.y;
  const int tid = threadIdx.x;
  const int d = blockIdx.x * 128 + tid;
  const int s_base = seg * (Ss / NSEG);
  const float* __restrict__ arow = attn + (size_t)b * Mm * Ss;
  float acc[Mm];
#pragma unroll
  for (int m = 0; m < Mm; ++m) acc[m] = 0.f;

  for (int tile = 0; tile < (Ss / NSEG) / 128; ++tile) {
    const int s0base = s_base + tile * 128;
    __syncthreads();
#pragma unroll
    for (int r = 0; r < Mm; ++r)
      att[r * 128 + tid] = arow[(size_t)r * Ss + s0base + tid];
    __syncthreads();
    const float* xp =
        tokens + ((size_t)b * (Ss + 1) + 1 + (size_t)s0base) * Dd + d;
#pragma unroll 4
    for (int s0 = 0; s0 < 128; ++s0) {
      const float xv = xp[(size_t)s0 * Dd];
#pragma unroll
      for (int m = 0; m < Mm; ++m) acc[m] = fmaf(att[m * 128 + s0], xv, acc[m]);
    }
  }
#pragma unroll
  for (int m = 0; m < Mm; ++m)
    merged_part[(((size_t)seg * Bb + b) * Mm + m) * Dd + d] = acc[m];
}

// =====================================================================
// Kernel 5b: reduce merged partials into pooled_learn head. grid (B).
// =====================================================================
__global__ __launch_bounds__(256) void k_merged_reduce(
    const float* __restrict__ merged_part, float* __restrict__ pooled_learn) {
  const int b = blockIdx.x;
  const int tid = threadIdx.x;
#pragma unroll
  for (int m = 0; m < Mm; ++m) {
    for (int r = 0; r < 3; ++r) {
      const int col = tid + r * 256;
      float v = 0.f;
#pragma unroll
      for (int seg = 0; seg < NSEG; ++seg)
        v += merged_part[(((size_t)seg * Bb + b) * Mm + m) * Dd + col];
      pooled_learn[(size_t)b * ((Mm + 1) * Dd) + m * Dd + col] = v;
    }
  }
}

// =====================================================================
// Kernel 6: WMMA f32 GEMM, D = A(16xKK) @ Bw(KKxN) + bias, optional GELU.
// Split-K: 8 waves per block, each wave WMMAs its KK/8 slice into a
// private 16x16 tile (V_WMMA_F32_16X16X4_F32); partial tiles are summed
// in LDS, then bias/GELU applied.
// KK is a compile-time template parameter so the inner loop has a
// *scalar, compile-time* trip count (KK/32 per wave after unroll-4) —
// this keeps loop control in SALU (no exec-mask loop) and lets the
// compiler batch the loads and stagger s_wait_loadcnt across 4 WMMAs.
//   A frag (2 VGPR): lanes 0-15 -> K+0/K+1, lanes 16-31 -> K+2/K+3
//   B frag symmetric; C/D: VGPR r holds row r (lanes 0-15) / r+8 (16-31)
// =====================================================================
template <bool GELU, int KK>
__global__ __launch_bounds__(256) void k_gemm16(
    const float* __restrict__ A, const float* __restrict__ Bw,
    const float* __restrict__ bias, float* __restrict__ out, int ldo) {
  constexpr int LDA = KK;        // A is 16 x KK row-major
  constexpr int LDB = Dd;        // Bw is KK x 768 row-major
  constexpr int KSLICE = KK / 8; // per-wave K slice (multiple of 4)
  __shared__ float sh[8 * 256];  // 8 partial 16x16 tiles
  const int tid = threadIdx.x;
  const int lane = tid & 31;
  const int wv = tid >> 5;
  const int n0 = blockIdx.x * 16;
  const int nn = lane & 15;
  const int koff = (lane >> 4) << 1;  // 0 for lanes 0-15, 2 for 16-31
  const int m = lane & 15;
  const int kbeg = wv * KSLICE;

  v8f acc = {};
  const float* __restrict__ arow = A + (size_t)m * LDA + kbeg + koff;
  const float* __restrict__ brow = Bw + (size_t)(kbeg + koff) * LDB + n0 + nn;
#pragma unroll 4
  for (int it = 0; it < KSLICE / 4; ++it) {  // scalar compile-time trip count
    float2 av = *(const float2*)(arow + it * 4);
    v2f a;
    a.x = av.x;
    a.y = av.y;
    v2f bfrag;
    bfrag.x = brow[(size_t)(it * 4) * LDB];
    bfrag.y = brow[(size_t)(it * 4 + 1) * LDB];
    acc = __builtin_amdgcn_wmma_f32_16x16x4_f32(false, a, false, bfrag,
                                                (short)0, acc, false, false);
  }
#pragma unroll
  for (int r = 0; r < 8; ++r) sh[wv * 256 + r * 32 + lane] = acc[r];
  __syncthreads();

  // element position p = tid maps to (r = p>>5, lane' = p&31):
  //   row = r + (lane' & 16 ? 8 : 0), col = n0 + (lane' & 15)
  float v = sh[tid];
#pragma unroll
  for (int w2 = 1; w2 < 8; ++w2) v += sh[w2 * 256 + tid];
  const int lp = tid & 31;
  const int row = (tid >> 5) + ((lp & 16) ? 8 : 0);
  const int col = n0 + (lp & 15);
  v += bias[col];
  if (GELU) v = gelu_exact(v);
  out[(size_t)row * ldo + col] = v;
}

// =====================================================================
// Launcher
// =====================================================================
extern "C" void kernel_launch(void* const* d_in, const int* in_sizes, int n_in,
                              void* d_out, int out_size, void* d_ws,
                              size_t ws_size, hipStream_t stream) {
  (void)in_sizes; (void)n_in; (void)out_size; (void)ws_size;
  const float* tokens = (const float*)d_in[0];
  const long long* lengths = (const long long*)d_in[1];
  const float* q = (const float*)d_in[2];
  const float* lns = (const float*)d_in[3];
  const float* lnb = (const float*)d_in[4];
  const float* w1a = (const float*)d_in[5];
  const float* b1a = (const float*)d_in[6];
  const float* w2a = (const float*)d_in[7];
  const float* b2a = (const float*)d_in[8];
  const float* w1b = (const float*)d_in[9];
  const float* b1b = (const float*)d_in[10];
  const float* w2b = (const float*)d_in[11];
  const float* b2b = (const float*)d_in[12];
  float* out = (float*)d_out;

  // workspace layout (floats)
  float* ws = (float*)d_ws;
  float* scores = ws;                                   // B*M*S     = 262144
  float* psum = scores + (size_t)Bb * Mm * Ss;          // B*16*D    = 196608
  float* pmax = psum + (size_t)Bb * NCHUNK * Dd;        //             196608
  float* pmin = pmax + (size_t)Bb * NCHUNK * Dd;        //             196608
  float* pooled_trad = pmin + (size_t)Bb * NCHUNK * Dd;     // B*3D  = 36864
  float* h_a = pooled_trad + (size_t)Bb * 3 * Dd;           // B*D   = 12288
  float* pooled_learn = h_a + (size_t)Bb * Dd;              // B*9D  = 110592
  float* h_b = pooled_learn + (size_t)Bb * (Mm + 1) * Dd;   // B*D   = 12288
  float* merged_part = h_b + (size_t)Bb * Dd;  // NSEG*B*M*D = 786432 (~7MB tot)

  // 1) fused pooling partials + LN + scores (streams tokens once)
  k_pool_scores<<<dim3(NCHUNK, Bb), 256, 0, stream>>>(
      tokens, lengths, q, lns, lnb, scores, psum, pmax, pmin);
  // 2) finalize mean/max/min pooling
  k_reduce_pool<<<Bb, 256, 0, stream>>>(psum, pmax, pmin, lengths, pooled_trad);
  // 3) masked softmax over S (in place)
  k_softmax<<<Bb * Mm, 256, 0, stream>>>(scores, lengths);
  // 4) clf tail of pooled_learn
  k_clf_copy<<<Bb, 256, 0, stream>>>(tokens, pooled_learn);
  // 5) merged = attn @ x (x from L2), split over S, then reduced
  k_merged<<<dim3(Dd / 128, NSEG, Bb), 128, 0, stream>>>(tokens, scores,
                                                         merged_part);
  k_merged_reduce<<<Bb, 256, 0, stream>>>(merged_part, pooled_learn);
  // 6) MLP A: h_a = gelu(pooled_trad @ w1a + b1a); out_fixed = h_a @ w2a + b2a
  k_gemm16<true, 3 * Dd><<<Dd / 16, 256, 0, stream>>>(pooled_trad, w1a, b1a,
                                                      h_a, Dd);
  k_gemm16<false, Dd><<<Dd / 16, 256, 0, stream>>>(h_a, w2a, b2a, out, 2 * Dd);
  // 7) MLP B: h_b = gelu(pooled_learn @ w1b + b1b); out_learn = h_b @ w2b + b2b
  k_gemm16<true, (Mm + 1) * Dd><<<Dd / 16, 256, 0, stream>>>(pooled_learn, w1b,
                                                             b1b, h_b, Dd);
  k_gemm16<false, Dd><<<Dd / 16, 256, 0, stream>>>(h_b, w2b, b2b, out + Dd,
                                                   2 * Dd);
}